// GMN_7189775254164
// MI455X (gfx1250) — compile-verified
//
#include <hip/hip_runtime.h>
#include <hip/hip_bf16.h>
#include <math.h>

#define VOCAB   21128
#define DIM     128
#define NCOL    20      // 10 conv + 10 cross
#define NPAD    32      // padded for two 16-wide WMMA N tiles
#define NN      50      // nodes per graph
#define EE      256     // edges per graph
#define EPS     1e-8f
#define MTILES  2       // M tiles (of 16 rows) per wave

typedef __attribute__((ext_vector_type(2))) float v2f;
typedef __attribute__((ext_vector_type(8))) float v8f;

// ---------------------------------------------------------------------------
// Kernel 1: T[v, 0:20] = emb[v,:] @ [conv_w | cross_w]   (exact f32 via WMMA)
// One wave32 per 32-row vocab tile (2 M-tiles). K=128 swept in 32 steps of 4
// using V_WMMA_F32_16X16X4_F32; B fetched once per k-step, shared by 4 WMMAs.
// Out-of-range rows are CLAMPED (not masked): they only affect their own D
// rows, whose stores are guarded — avoids per-load exec predication.
// ---------------------------------------------------------------------------
__global__ __launch_bounds__(32) void token_xw_kernel(
    const float* __restrict__ emb,      // [VOCAB,128]
    const float* __restrict__ conv_w,   // [128,10]
    const float* __restrict__ cross_w,  // [128,10]
    float* __restrict__ T)              // [VOCAB,20]
{
    __shared__ float Wp[DIM][NPAD];     // zero-padded weight panel
    const int lane = threadIdx.x;       // 0..31 (wave32)

    for (int idx = lane; idx < DIM * NPAD; idx += 32) {
        int k = idx / NPAD, c = idx % NPAD;
        float v = 0.0f;
        if (c < 10)        v = conv_w [k * 10 + c];
        else if (c < NCOL) v = cross_w[k * 10 + (c - 10)];
        Wp[k][c] = v;
    }
    __syncthreads();

    const int m0   = blockIdx.x * (16 * MTILES);
    const int half = lane >> 4;         // 0: lanes 0-15, 1: lanes 16-31
    const int l15  = lane & 15;

    const float* __restrict__ arow[MTILES];
    #pragma unroll
    for (int t = 0; t < MTILES; ++t) {
        int r = m0 + t * 16 + l15;
        if (r > VOCAB - 1) r = VOCAB - 1;           // clamp: safe, finite data
        arow[t] = emb + (size_t)r * DIM;
    }

    v8f acc0[MTILES], acc1[MTILES];
    #pragma unroll
    for (int t = 0; t < MTILES; ++t) { acc0[t] = (v8f){}; acc1[t] = (v8f){}; }

    for (int k0 = 0; k0 < DIM; k0 += 4) {
        const int ka = k0 + half * 2;   // A/B: VGPR0 holds K=k0 (lo lanes), K=k0+2 (hi lanes)
        v2f b0, b1;                     // B: N = lane&15 (+16 for tile 1)
        b0.x = Wp[ka][l15];       b0.y = Wp[ka + 1][l15];
        b1.x = Wp[ka][l15 + 16];  b1.y = Wp[ka + 1][l15 + 16];
        #pragma unroll
        for (int t = 0; t < MTILES; ++t) {
            const v2f a = *(const v2f*)(arow[t] + ka);   // one b64 load, no predication
            acc0[t] = __builtin_amdgcn_wmma_f32_16x16x4_f32(false, a, false, b0, (short)0, acc0[t], false, false);
            acc1[t] = __builtin_amdgcn_wmma_f32_16x16x4_f32(false, a, false, b1, (short)0, acc1[t], false, false);
        }
    }

    // C/D layout: VGPR r -> row M = r + half*8, column N = lane&15 (tile 0) / +16 (tile 1)
    #pragma unroll
    for (int t = 0; t < MTILES; ++t) {
        #pragma unroll
        for (int r = 0; r < 8; ++r) {
            const int row = m0 + t * 16 + r + half * 8;
            if (row < VOCAB) {
                T[(size_t)row * NCOL + l15] = acc0[t][r];       // cols 0..15
                const int col1 = l15 + 16;
                if (col1 < NCOL)
                    T[(size_t)row * NCOL + col1] = acc1[t][r];  // cols 16..19
            }
        }
    }
}

// ---------------------------------------------------------------------------
// Kernel 2: one block (128 threads) per sample; everything in LDS.
// Deterministic (no FP atomics): GCN aggregation via edge scans.
// ---------------------------------------------------------------------------
__global__ __launch_bounds__(128) void sample_kernel(
    const int*   __restrict__ u_tok,    // [B,50]
    const int*   __restrict__ r_tok,    // [B,50]
    const int*   __restrict__ u_edge,   // [B,2,256]
    const int*   __restrict__ r_edge,   // [B,2,256]
    const float* __restrict__ T,        // [VOCAB,20]
    const float* __restrict__ conv_b,   // [10]
    const float* __restrict__ attn,     // [50,50]
    const float* __restrict__ assign_w, // [50,10]
    const float* __restrict__ mp_w1,    // [11,20]
    const float* __restrict__ mp_b1,    // [20]
    const float* __restrict__ mp_w2,    // [20,20]
    const float* __restrict__ mp_b2,    // [20]
    const float* __restrict__ ffn_w1,   // [80,40]
    const float* __restrict__ ffn_b1,   // [40]
    const float* __restrict__ ffn_w2,   // [40,1]
    const float* __restrict__ ffn_b2,   // [1]
    float* __restrict__ out)            // [B,1]
{
    __shared__ float Tu[NN][NCOL];
    __shared__ float Tr[NN][NCOL];
    __shared__ int   eds[2][2][EE];     // [graph][src/dst][e]
    __shared__ float dinv[2][NN];
    __shared__ float selfF[2][NN][10];
    __shared__ float crossF[2][NN][10];
    __shared__ float dist[2][NN];
    __shared__ float h1[2][NN][20];
    __shared__ float hh[2][NN][20];
    __shared__ float gmax[2][20];
    __shared__ float feat[80];
    __shared__ float f1[40];
    __shared__ float s_attn[NN * NN];

    const int tid = threadIdx.x;
    const int b   = blockIdx.x;

    // ---- gather T rows for both token sequences; load edges; load attn ----
    for (int i = tid; i < NN * NCOL; i += 128) {
        const int n = i / NCOL, c = i % NCOL;
        Tu[n][c] = T[(size_t)u_tok[(size_t)b * NN + n] * NCOL + c];
        Tr[n][c] = T[(size_t)r_tok[(size_t)b * NN + n] * NCOL + c];
    }
    for (int i = tid; i < 2 * 2 * EE; i += 128) {
        const int gr = i / (2 * EE), rem = i % (2 * EE);
        const int* ed = (gr ? r_edge : u_edge) + (size_t)b * 2 * EE;
        eds[gr][rem / EE][rem % EE] = ed[rem];
    }
    for (int i = tid; i < NN * NN; i += 128) s_attn[i] = attn[i];
    __syncthreads();

    // ---- degrees (incoming + self-loop) -> dinv ----
    for (int i = tid; i < 2 * NN; i += 128) {
        const int gr = i / NN, n = i % NN;
        int cnt = 1;                     // self loop
        for (int e = 0; e < EE; ++e) cnt += (eds[gr][1][e] == n);
        dinv[gr][n] = rsqrtf((float)cnt);
    }
    __syncthreads();

    // ---- GCN aggregation: selfF = D^-1/2 (A+I) D^-1/2 (xW) + b ----
    for (int i = tid; i < 2 * NN; i += 128) {
        const int gr = i / NN, n = i % NN;
        float (*Txw)[NCOL] = gr ? Tr : Tu;
        const float di = dinv[gr][n];
        float acc[10];
        #pragma unroll
        for (int c = 0; c < 10; ++c) acc[c] = 0.0f;
        for (int e = 0; e < EE; ++e) {
            if (eds[gr][1][e] == n) {
                const int s = eds[gr][0][e];
                const float w = dinv[gr][s];
                #pragma unroll
                for (int c = 0; c < 10; ++c) acc[c] += w * Txw[s][c];
            }
        }
        #pragma unroll
        for (int c = 0; c < 10; ++c)
            selfF[gr][n][c] = di * (acc[c] + di * Txw[n][c]) + conv_b[c];
    }
    __syncthreads();

    // ---- cross features: u_cross = attn @ (r xW_cross); r_cross symmetric ----
    for (int i = tid; i < 2 * NN * 10; i += 128) {
        const int gr = i / (NN * 10), rem = i % (NN * 10);
        const int n = rem / 10, c = rem % 10;
        float (*Tc)[NCOL] = gr ? Tu : Tr;   // note: cross uses the OTHER graph's emb
        float acc = 0.0f;
        for (int j = 0; j < NN; ++j) acc += s_attn[n * NN + j] * Tc[j][10 + c];
        crossF[gr][n][c] = acc;
    }
    __syncthreads();

    // ---- cosine distance per node ----
    for (int i = tid; i < 2 * NN; i += 128) {
        const int gr = i / NN, n = i % NN;
        float dot = 0.0f, na = 0.0f, nc = 0.0f;
        #pragma unroll
        for (int c = 0; c < 10; ++c) {
            const float aw = assign_w[n * 10 + c];
            const float a  = aw * selfF[gr][n][c];
            const float cc = aw * crossF[gr][n][c];
            dot += a * cc; na += a * a; nc += cc * cc;
        }
        na = fmaxf(sqrtf(na), EPS);
        nc = fmaxf(sqrtf(nc), EPS);
        dist[gr][n] = dot / (na * nc);
    }
    __syncthreads();

    // ---- match MLP layer 1: h1 = relu([dist, selfF] @ w1 + b1) ----
    for (int i = tid; i < 2 * NN * 20; i += 128) {
        const int gr = i / (NN * 20), rem = i % (NN * 20);
        const int n = rem / 20, j = rem % 20;
        float acc = mp_b1[j] + dist[gr][n] * mp_w1[0 * 20 + j];
        #pragma unroll
        for (int k = 0; k < 10; ++k) acc += selfF[gr][n][k] * mp_w1[(k + 1) * 20 + j];
        h1[gr][n][j] = fmaxf(acc, 0.0f);
    }
    __syncthreads();

    // ---- match MLP layer 2: hh = h1 @ w2 + b2 ----
    for (int i = tid; i < 2 * NN * 20; i += 128) {
        const int gr = i / (NN * 20), rem = i % (NN * 20);
        const int n = rem / 20, j = rem % 20;
        float acc = mp_b2[j];
        #pragma unroll
        for (int k = 0; k < 20; ++k) acc += h1[gr][n][k] * mp_w2[k * 20 + j];
        hh[gr][n][j] = acc;
    }
    __syncthreads();

    // ---- max-pool over nodes ----
    for (int i = tid; i < 2 * 20; i += 128) {
        const int gr = i / 20, j = i % 20;
        float m = -INFINITY;
        for (int n = 0; n < NN; ++n) m = fmaxf(m, hh[gr][n][j]);
        gmax[gr][j] = m;
    }
    __syncthreads();

    // ---- feature vector [g_u, g_r, g_u*g_r, |g_u-g_r|] ----
    for (int i = tid; i < 80; i += 128) {
        const int j = i % 20, sec = i / 20;
        const float gu = gmax[0][j], gr_ = gmax[1][j];
        float v;
        if      (sec == 0) v = gu;
        else if (sec == 1) v = gr_;
        else if (sec == 2) v = gu * gr_;
        else               v = fabsf(gu - gr_);
        feat[i] = v;
    }
    __syncthreads();

    // ---- FFN layer 1 ----
    for (int i = tid; i < 40; i += 128) {
        float acc = ffn_b1[i];
        for (int k = 0; k < 80; ++k) acc += feat[k] * ffn_w1[k * 40 + i];
        f1[i] = fmaxf(acc, 0.0f);
    }
    __syncthreads();

    // ---- FFN layer 2 + sigmoid ----
    if (tid == 0) {
        float acc = ffn_b2[0];
        for (int k = 0; k < 40; ++k) acc += f1[k] * ffn_w2[k];
        out[b] = 1.0f / (1.0f + expf(-acc));
    }
}

// ---------------------------------------------------------------------------
extern "C" void kernel_launch(void* const* d_in, const int* in_sizes, int n_in,
                              void* d_out, int out_size, void* d_ws, size_t ws_size,
                              hipStream_t stream) {
    const int*   u_tok    = (const int*)  d_in[0];
    const int*   r_tok    = (const int*)  d_in[1];
    const int*   u_edge   = (const int*)  d_in[2];
    const int*   r_edge   = (const int*)  d_in[3];
    const float* emb      = (const float*)d_in[4];
    const float* conv_w   = (const float*)d_in[5];
    const float* conv_b   = (const float*)d_in[6];
    const float* cross_w  = (const float*)d_in[7];
    const float* attn     = (const float*)d_in[8];
    const float* assign_w = (const float*)d_in[9];
    const float* mp_w1    = (const float*)d_in[10];
    const float* mp_b1    = (const float*)d_in[11];
    const float* mp_w2    = (const float*)d_in[12];
    const float* mp_b2    = (const float*)d_in[13];
    const float* ffn_w1   = (const float*)d_in[14];
    const float* ffn_b1   = (const float*)d_in[15];
    const float* ffn_w2   = (const float*)d_in[16];
    const float* ffn_b2   = (const float*)d_in[17];
    float* out = (float*)d_out;

    float* T = (float*)d_ws;            // [VOCAB,20] = 1.69 MB scratch
    const int B = in_sizes[0] / NN;

    const int rows_per_block = 16 * MTILES;
    token_xw_kernel<<<(VOCAB + rows_per_block - 1) / rows_per_block, 32, 0, stream>>>(
        emb, conv_w, cross_w, T);
    sample_kernel<<<B, 128, 0, stream>>>(u_tok, r_tok, u_edge, r_edge, T,
                                         conv_b, attn, assign_w,
                                         mp_w1, mp_b1, mp_w2, mp_b2,
                                         ffn_w1, ffn_b1, ffn_w2, ffn_b2, out);
}